// ModelWrapper_25443386262295
// MI455X (gfx1250) — compile-verified
//
#include <hip/hip_runtime.h>
#include <cstdint>

#define NBOX 1572864
#define KTOP 4096
#define NBINS 262144      // float bits >> 14 (18 bits)
#define CAP   8192        // compaction capacity (2x K)

typedef __attribute__((ext_vector_type(2))) float v2f;
typedef __attribute__((ext_vector_type(8))) float v8f;
typedef __attribute__((ext_vector_type(4))) unsigned int v4u;
typedef __attribute__((ext_vector_type(8))) int v8i;
typedef __attribute__((ext_vector_type(4))) int v4i;

// ---------------------------------------------------------------- init
__global__ void init_kernel(unsigned* hist, unsigned* meta, unsigned long long* ckeys) {
    const int i = blockIdx.x * blockDim.x + threadIdx.x;
    const int stride = gridDim.x * blockDim.x;
    for (int t = i; t < NBINS; t += stride) hist[t] = 0u;
    if (i < 16) meta[i] = 0u;
    if (i < CAP) ckeys[i] = 0ull;
}

// ------------------------------------------------- decode + score + hist
__global__ __launch_bounds__(256)
void decode_kernel(const float* __restrict__ cls, const float* __restrict__ reg,
                   const float* __restrict__ anc, float* __restrict__ boxes,
                   float* __restrict__ mscore, unsigned* __restrict__ hist) {
    const int i = blockIdx.x * blockDim.x + threadIdx.x;
    if (i >= NBOX) return;
    const float4 a = reinterpret_cast<const float4*>(anc)[i];
    const float4 r = reinterpret_cast<const float4*>(reg)[i];
    const float d0 = r.x * 0.1f, d1 = r.y * 0.1f, d2 = r.z * 0.2f, d3 = r.w * 0.2f;
    const float w = a.z - a.x, h = a.w - a.y;
    const float cx = a.x + 0.5f * w, cy = a.y + 0.5f * h;
    const float pcx = cx + d0 * w, pcy = cy + d1 * h;
    const float pw = __expf(d2) * w, ph = __expf(d3) * h;
    float4 b;
    b.x = fminf(fmaxf(pcx - 0.5f * pw, 0.f), 1024.f);
    b.y = fminf(fmaxf(pcy - 0.5f * ph, 0.f), 1024.f);
    b.z = fminf(fmaxf(pcx + 0.5f * pw, 0.f), 1024.f);
    b.w = fminf(fmaxf(pcy + 0.5f * ph, 0.f), 1024.f);
    reinterpret_cast<float4*>(boxes)[i] = b;
    float s = cls[i * 2 + 1];
    s = (s > 0.5f) ? s : 0.f;
    mscore[i] = s;
    if (s > 0.f) atomicAdd(&hist[__float_as_uint(s) >> 14], 1u);
}

// -------------------------------------------- find the K-th score's bin
__global__ __launch_bounds__(512)
void cutoff_kernel(const unsigned* __restrict__ hist, unsigned* __restrict__ meta) {
    __shared__ unsigned csum[512];
    const int tid = threadIdx.x;
    unsigned s = 0;
    const int base = tid * 512;
    for (int b = 0; b < 512; ++b) s += hist[base + b];
    csum[tid] = s;
    __syncthreads();
    if (tid == 0) {
        unsigned acc = 0;
        int ch = 0;
        for (int c = 511; c >= 0; --c) {
            if (acc + csum[c] >= (unsigned)KTOP) { ch = c; break; }
            acc += csum[c];
        }
        unsigned cut = (unsigned)(ch * 512);
        for (int b = ch * 512 + 511; b >= ch * 512; --b) {
            acc += hist[b];
            if (acc >= (unsigned)KTOP) { cut = (unsigned)b; break; }
        }
        meta[0] = cut;
    }
}

// ------------------------------------------------------------ compaction
__global__ __launch_bounds__(256)
void compact_kernel(const float* __restrict__ mscore, unsigned* __restrict__ meta,
                    unsigned long long* __restrict__ ckeys) {
    const int i = blockIdx.x * blockDim.x + threadIdx.x;
    if (i >= NBOX) return;
    const float s = mscore[i];
    if (s <= 0.f) return;
    const unsigned bits = __float_as_uint(s);
    if ((bits >> 14) >= meta[0]) {
        const unsigned pos = atomicAdd(&meta[1], 1u);
        if (pos < (unsigned)CAP)
            ckeys[pos] = ((unsigned long long)bits << 32) | (unsigned)(~i);
    }
}

// ------------------------------- one-block bitonic sort of 8192 u64 keys
__global__ __launch_bounds__(1024)
void sort_kernel(const unsigned long long* __restrict__ ckeys,
                 float* __restrict__ tops, unsigned* __restrict__ order) {
    __shared__ unsigned long long s[CAP];   // 64 KB
    const int tid = threadIdx.x;
    for (int t = tid; t < CAP; t += 1024) s[t] = ckeys[t];
    __syncthreads();
    for (unsigned k = 2; k <= (unsigned)CAP; k <<= 1) {
        for (unsigned j = k >> 1; j > 0; j >>= 1) {
            for (unsigned t = tid; t < CAP / 2; t += 1024) {
                const unsigned i = 2u * j * (t / j) + (t % j);
                const unsigned l = i ^ j;
                const unsigned long long a = s[i], b = s[l];
                const bool dir = ((i & k) == 0u);     // dir==true -> descending
                if (dir ? (a < b) : (a > b)) { s[i] = b; s[l] = a; }
            }
            __syncthreads();
        }
    }
    for (int t = tid; t < KTOP; t += 1024) {
        const unsigned long long key = s[t];
        float sc = __uint_as_float((unsigned)(key >> 32));
        unsigned idx = ~(unsigned)key;
        if (key == 0ull) { sc = 0.f; idx = 0u; }
        tops[t] = sc;
        order[t] = idx;
    }
}

// ---------------------------------------------------- gather + areas
__global__ __launch_bounds__(256)
void gather_kernel(const float* __restrict__ boxes, const unsigned* __restrict__ order,
                   float* __restrict__ cand, float* __restrict__ areas) {
    const int k = blockIdx.x * blockDim.x + threadIdx.x;
    if (k >= KTOP) return;
    const unsigned j = order[k];
    const float4 b = reinterpret_cast<const float4*>(boxes)[j];
    reinterpret_cast<float4*>(cand)[k] = b;
    areas[k] = (b.z - b.x) * (b.w - b.y);
}

// ------------- IoU suppression bit matrix; area_i+area_j via WMMA f32
__global__ __launch_bounds__(256)
void iou_mask_kernel(const float* __restrict__ cand, const float* __restrict__ areas,
                     unsigned* __restrict__ mask) {
    __shared__ float rb[64];
    __shared__ float ra[16];
    const int tid = threadIdx.x;
    const int rt = blockIdx.y;                      // row tile: 16 rows
    if (tid < 64) rb[tid] = cand[rt * 64 + tid];
    if (tid < 16) ra[tid] = areas[rt * 16 + tid];
    __syncthreads();

    const int lane = tid & 31;
    const int wave = tid >> 5;
    const int wordIdx = blockIdx.x * 8 + wave;      // 0..127 (32 columns each)
    const int nLoc = lane & 15;
    const int mBase = (lane >> 4) * 8;
    const bool lowHalf = (lane < 16);

    unsigned bits[2][8];
    #pragma unroll
    for (int t = 0; t < 2; ++t) {
        const int c0 = wordIdx * 32 + t * 16;
        const int col = c0 + nLoc;
        const float4 cb = reinterpret_cast<const float4*>(cand)[col];
        const float carea = areas[col];
        // A row m = [area_row[m], 1, 0, 0]; B col n = [1, area_col[n], 0, 0]^T
        // => D[m][n] = area_row[m] + area_col[n]  (rank-2 outer sum on the MMA unit)
        v2f a, b;
        a.x = lowHalf ? ra[nLoc] : 0.f;   // K=0 (lanes 0-15 hold K=0,1)
        a.y = lowHalf ? 1.f : 0.f;        // K=1
        b.x = lowHalf ? 1.f : 0.f;        // K=0 row of B
        b.y = lowHalf ? carea : 0.f;      // K=1 row of B
        v8f c = {};
        v8f sum = __builtin_amdgcn_wmma_f32_16x16x4_f32(
            false, a, false, b, (short)0, c, false, false);
        #pragma unroll
        for (int v = 0; v < 8; ++v) {
            const int ml = mBase + v;
            const float rx1 = rb[ml * 4 + 0], ry1 = rb[ml * 4 + 1];
            const float rx2 = rb[ml * 4 + 2], ry2 = rb[ml * 4 + 3];
            float iw = fminf(rx2, cb.z) - fmaxf(rx1, cb.x);
            float ih = fminf(ry2, cb.w) - fmaxf(ry1, cb.y);
            iw = fmaxf(iw, 0.f);
            ih = fmaxf(ih, 0.f);
            const float inter = iw * ih;
            const float denom = sum[v] - inter + 1e-8f;   // a_i + a_j - inter + eps
            const int mg = rt * 16 + ml;
            const bool pred = (inter > 0.1f * denom) && (col > mg);
            bits[t][v] = (unsigned)__ballot(pred);
        }
    }
    if (lane < 16) {
        const int v = lane & 7;
        const int hi = lane >> 3;
        const unsigned h0 = hi ? (bits[0][v] >> 16) : (bits[0][v] & 0xFFFFu);
        const unsigned h1 = hi ? (bits[1][v] >> 16) : (bits[1][v] & 0xFFFFu);
        const int row = rt * 16 + hi * 8 + v;
        mask[row * 128 + wordIdx] = h0 | (h1 << 16);
    }
}

// ---------------- TDM: async-load one 32-row x 128-word mask chunk to LDS
__device__ __forceinline__ void tdm_load_chunk(const unsigned* gbase, unsigned ldsAddr) {
    const unsigned long long ga = (unsigned long long)(uintptr_t)gbase;
    v4u g0;
    g0.x = 1u;                                        // count=1, user descriptor
    g0.y = ldsAddr;                                   // LDS byte address
    g0.z = (unsigned)(ga & 0xFFFFFFFFull);            // global_addr[31:0]
    g0.w = (unsigned)((ga >> 32) & 0x01FFFFFFull) | (2u << 30);  // addr[56:32] | type=2
    v8i g1;
    g1[0] = 0x20000;                  // workgroup_mask=0, data_size=4B
    g1[1] = (int)(128u << 16);        // tensor_dim0 = 128 dwords/row (low16)
    g1[2] = (int)(4096u << 16);       // tensor_dim1 = 4096 rows (low16)
    g1[3] = (int)(128u << 16);        // tile_dim0 = 128
    g1[4] = 32;                       // tile_dim1 = 32 rows, tile_dim2 = 0
    g1[5] = 128;                      // tensor_dim0_stride = 128 (low32)
    g1[6] = 0;
    g1[7] = 0;
    const v4i z4 = {0, 0, 0, 0};
    const v8i z8 = {0, 0, 0, 0, 0, 0, 0, 0};
    __builtin_amdgcn_tensor_load_to_lds(g0, g1, z4, z4, z8, 0);
}

// ---------------------------- serial NMS scan, TDM double-buffered chunks
__global__ __launch_bounds__(128)
void nms_scan_kernel(const unsigned* __restrict__ mask, const float* __restrict__ tops,
                     unsigned* __restrict__ keepw) {
    __shared__ unsigned removed[128];
    __shared__ unsigned buf[2][32 * 128];             // 2 x 16 KB chunk buffers
    const int tid = threadIdx.x;
    removed[tid] = 0u;
    __syncthreads();
    for (int i = tid; i < KTOP; i += 128)
        if (tops[i] <= 0.f) atomicOr(&removed[i >> 5], 1u << (i & 31));
    __syncthreads();

    if (tid < 32) tdm_load_chunk(mask, (unsigned)(uintptr_t)&buf[0][0]);

    for (int c = 0; c < 128; ++c) {
        if (tid < 32) {
            if (c + 1 < 128) {
                tdm_load_chunk(mask + (c + 1) * 32 * 128,
                               (unsigned)(uintptr_t)&buf[(c + 1) & 1][0]);
                __builtin_amdgcn_s_wait_tensorcnt(1);  // chunk c done (in-order)
            } else {
                __builtin_amdgcn_s_wait_tensorcnt(0);
            }
        }
        __syncthreads();
        const unsigned* rowbuf = &buf[c & 1][0];
        for (int r = 0; r < 32; ++r) {
            const int i = c * 32 + r;
            const bool k = ((removed[i >> 5] >> (i & 31)) & 1u) == 0u;
            __syncthreads();
            if (k) removed[tid] |= rowbuf[r * 128 + tid];
            __syncthreads();
        }
    }
    keepw[tid] = ~removed[tid];
}

// --------------------------------------------------------------- output
__global__ __launch_bounds__(256)
void output_kernel(const float* __restrict__ cand, const float* __restrict__ tops,
                   const unsigned* __restrict__ keepw, const int* __restrict__ Hp,
                   const int* __restrict__ Wp, float* __restrict__ out) {
    const int k = blockIdx.x * blockDim.x + threadIdx.x;
    if (k >= KTOP) return;
    const float kf = ((keepw[k >> 5] >> (k & 31)) & 1u) ? 1.f : 0.f;
    const float sx = (float)Wp[0] * (1.f / 1024.f);
    const float sy = (float)Hp[0] * (1.f / 1024.f);
    const float4 b = reinterpret_cast<const float4*>(cand)[k];
    out[k * 5 + 0] = b.x * sx * kf;
    out[k * 5 + 1] = b.y * sy * kf;
    out[k * 5 + 2] = b.z * sx * kf;
    out[k * 5 + 3] = b.w * sy * kf;
    out[k * 5 + 4] = tops[k] * kf;
}

extern "C" void kernel_launch(void* const* d_in, const int* in_sizes, int n_in,
                              void* d_out, int out_size, void* d_ws, size_t ws_size,
                              hipStream_t stream) {
    const float* cls = (const float*)d_in[0];   // (1,N,2)
    const float* reg = (const float*)d_in[1];   // (1,N,4)
    const float* anc = (const float*)d_in[2];   // (1,N,4)
    const int* Hp = (const int*)d_in[3];
    const int* Wp = (const int*)d_in[4];
    float* out = (float*)d_out;

    char* ws = (char*)d_ws;
    float*              boxes  = (float*)(ws);                       // 25165824 B
    float*              mscore = (float*)(ws + 25165824);            //  6291456 B
    unsigned*           hist   = (unsigned*)(ws + 31457280);         //  1048576 B
    unsigned*           meta   = (unsigned*)(ws + 32505856);         //      256 B
    unsigned long long* ckeys  = (unsigned long long*)(ws + 32506112); //  65536 B
    float*              tops   = (float*)(ws + 32571648);            //    16384 B
    unsigned*           order  = (unsigned*)(ws + 32588032);         //    16384 B
    float*              cand   = (float*)(ws + 32604416);            //    65536 B
    float*              areas  = (float*)(ws + 32669952);            //    16384 B
    unsigned*           mask   = (unsigned*)(ws + 32686336);         //  2097152 B
    unsigned*           keepw  = (unsigned*)(ws + 34783488);         //      512 B

    init_kernel<<<256, 256, 0, stream>>>(hist, meta, ckeys);
    decode_kernel<<<NBOX / 256, 256, 0, stream>>>(cls, reg, anc, boxes, mscore, hist);
    cutoff_kernel<<<1, 512, 0, stream>>>(hist, meta);
    compact_kernel<<<NBOX / 256, 256, 0, stream>>>(mscore, meta, ckeys);
    sort_kernel<<<1, 1024, 0, stream>>>(ckeys, tops, order);
    gather_kernel<<<KTOP / 256, 256, 0, stream>>>(boxes, order, cand, areas);
    iou_mask_kernel<<<dim3(16, 256, 1), 256, 0, stream>>>(cand, areas, mask);
    nms_scan_kernel<<<1, 128, 0, stream>>>(mask, tops, keepw);
    output_kernel<<<KTOP / 256, 256, 0, stream>>>(cand, tops, keepw, Hp, Wp, out);
}